// shifts_24275155157677
// MI455X (gfx1250) — compile-verified
//
#include <hip/hip_runtime.h>

// shifts: out[b, k*C+c, h, w] = x[b, c, h-sx_k, w-sy_k] (zero-padded), k in 9 offsets
// B=16 C=32 H=192 W=192 -> out (16, 288, 192, 192) f32.
// Pure data movement: ~75.5MB read, ~679.5MB write -> store-bound, ~29us floor @23.3TB/s.
// gfx1250 path: TDM tensor_load_to_lds stages a (TILE_H+2)xW input tile per block,
// 128-bit non-temporal global stores stream the 9 shifted copies.

typedef float v4f __attribute__((ext_vector_type(4)));
typedef unsigned int u32x4 __attribute__((ext_vector_type(4)));
typedef int i32x8 __attribute__((ext_vector_type(8)));
typedef int i32x4 __attribute__((ext_vector_type(4)));

#define NB 16
#define NC 32
#define NH 192
#define NW 192
#define TILE_H 32
#define NTHREADS 256

__global__ __launch_bounds__(NTHREADS) void shifts_24275155157677_kernel(
    const float* __restrict__ x, float* __restrict__ out) {
  __shared__ __align__(16) float buf[(TILE_H + 2) * NW];  // 34*192 floats = 25.5 KB

  const int htile = blockIdx.x;   // 0..5
  const int c     = blockIdx.y;   // 0..31
  const int b     = blockIdx.z;   // 0..15
  const int h0    = htile * TILE_H;
  const int t     = threadIdx.x;

  // Zero the two potential halo rows (row 0 <-> input h0-1, row TILE_H+1 <-> h0+TILE_H).
  // The TDM overwrites whichever of them is actually in-bounds.
  for (int i = t; i < 2 * NW; i += NTHREADS) {
    int row = (i < NW) ? 0 : (TILE_H + 1);
    int col = (i < NW) ? i : (i - NW);
    buf[row * NW + col] = 0.0f;
  }
  __syncthreads();

  // One TDM descriptor per block: DMA input rows [lo, hi] of plane (b,c) into LDS.
  if (t == 0) {
    int lo = h0 - 1;        if (lo < 0) lo = 0;
    int hi = h0 + TILE_H;   if (hi > NH - 1) hi = NH - 1;
    int n  = hi - lo + 1;   // TILE_H+1 or TILE_H+2 rows

    unsigned long long ga =
        (unsigned long long)(uintptr_t)(x + (((size_t)(b * NC + c) * NH + lo) * NW));
    unsigned int lds_off =
        (unsigned int)(uintptr_t)(&buf[(lo - (h0 - 1)) * NW]);  // low 32b = LDS byte offset

    // ---- D# group 0 (128b): count=1 | lds_addr | global_addr | type=2 ----
    u32x4 g0;
    g0.x = 1u;                                              // count=1, is_restore=0, gather=0
    g0.y = lds_off;                                         // lds_addr [63:32]
    g0.z = (unsigned int)(ga & 0xFFFFFFFFull);              // global_addr [95:64]
    g0.w = (unsigned int)((ga >> 32) & 0x01FFFFFFull)       // global_addr [120:96]
         | (2u << 30);                                      // type=2 ("image") [127:126]

    // ---- D# group 1 (256b) ----
    i32x8 g1;
    g1[0] = (int)(2u << 16);                                // data_size=2 (4 bytes); mask=0
    g1[1] = (int)(((unsigned)NW & 0xFFFFu) << 16);          // tensor_dim0[15:0] @ bits 63:48
    g1[2] = (int)((((unsigned)NW >> 16) & 0xFFFFu)          // tensor_dim0[31:16] @ 79:64
                | (((unsigned)n & 0xFFFFu) << 16));         // tensor_dim1[15:0] @ 111:80
    g1[3] = (int)((((unsigned)n >> 16) & 0xFFFFu)           // tensor_dim1[31:16]
                | (((unsigned)NW & 0xFFFFu) << 16));        // tile_dim0 @ 127:112
    g1[4] = (int)((unsigned)n & 0xFFFFu);                   // tile_dim1 = n; tile_dim2 = 0
    g1[5] = NW;                                             // tensor_dim0_stride[31:0]
    g1[6] = 0;                                              // td0_stride hi; td1_stride lo
    g1[7] = 0;                                              // td1_stride hi

    i32x4 g2 = {0, 0, 0, 0};                                // 2-D tensor: groups 2/3 unused
    i32x4 g3 = {0, 0, 0, 0};
    i32x8 g4 = {0, 0, 0, 0, 0, 0, 0, 0};                    // extra group (clang-23 form): zero

    __builtin_amdgcn_tensor_load_to_lds(g0, g1, g2, g3, g4, 0); // cpol=0
    __builtin_amdgcn_s_wait_tensorcnt(0);
  }
  __syncthreads();

  // Emit the 9 shifted copies. Offset order matches the reference:
  // (0,0), then (sx,sy) nested loop skipping (0,0).
  const int SX[9] = {0, -1, -1, -1, 0, 0, 1, 1, 1};
  const int SY[9] = {0, -1,  0,  1, -1, 1, -1, 0, 1};

  const int W4 = NW / 4;  // 48 float4 per row
#pragma unroll
  for (int k = 0; k < 9; ++k) {
    const int sx = SX[k];
    const int sy = SY[k];
    float* orow0 = out + (((size_t)(b * 9 + k) * NC + c) * NH + h0) * NW;
    for (int j = t; j < TILE_H * W4; j += NTHREADS) {
      const int r  = j / W4;
      const int w4 = j - r * W4;
      const float* lp = &buf[(r + 1 - sx) * NW];  // lds row in [0, TILE_H+1] always
      v4f v;
      if (sy == 0) {
        v = *(const v4f*)(lp + 4 * w4);           // ds_load_b128, conflict-free
      } else {
        const int c0 = 4 * w4 - sy;
#pragma unroll
        for (int e = 0; e < 4; ++e) {
          const int cc = c0 + e;
          v[e] = (cc >= 0 && cc < NW) ? lp[cc] : 0.0f;
        }
      }
      // write-once stream: non-temporal 128-bit store
      __builtin_nontemporal_store(v, (v4f*)(orow0 + (size_t)r * NW) + w4);
    }
  }
}

extern "C" void kernel_launch(void* const* d_in, const int* in_sizes, int n_in,
                              void* d_out, int out_size, void* d_ws, size_t ws_size,
                              hipStream_t stream) {
  (void)in_sizes; (void)n_in; (void)d_ws; (void)ws_size; (void)out_size;
  const float* x = (const float*)d_in[0];
  float* out = (float*)d_out;
  dim3 grid(NH / TILE_H, NC, NB);  // (6, 32, 16) blocks
  shifts_24275155157677_kernel<<<grid, NTHREADS, 0, stream>>>(x, out);
}